// PadToFullGraphEdgeEncoder_66271345377488
// MI455X (gfx1250) — compile-verified
//
#include <hip/hip_runtime.h>
#include <stdint.h>

// Problem constants (match reference)
#define B_G    128
#define N_PER  128
#define D_DIM  64
#define E_TOT  262144
#define EPG    (E_TOT / B_G)                 // 2048 edges per graph, contiguous
#define ROWS_PER_BLK 4                       // 128KB tile -> 2 workgroups/WGP
#define BLKS_PER_G   (N_PER / ROWS_PER_BLK)  // 32 row-blocks per graph
#define TILE_FLOATS  (ROWS_PER_BLK * N_PER * D_DIM)  // 32768 f32 = 128 KB LDS
#define M_TOT  (B_G * N_PER * N_PER)         // 2,097,152 (fits in int32)

typedef unsigned int u32x4 __attribute__((ext_vector_type(4)));
typedef int          i32x4 __attribute__((ext_vector_type(4)));
typedef int          i32x8 __attribute__((ext_vector_type(8)));

// ---------------------------------------------------------------------------
// Kernel 1: coalesced edge-index enumeration (rows then cols), written as f32
// (JAX promotion on concat of int32+float32 outputs -> float32; values
// <= 16383 are exact in f32). Pure 32-bit index math, float4 stores.
// M_TOT % 4 == 0, so the rows/cols branch is uniform within each float4.
// ---------------------------------------------------------------------------
extern "C" __global__ void idx_fill_kernel(float* __restrict__ out) {
  const int k4 = blockIdx.x * blockDim.x + threadIdx.x;
  const int k = k4 * 4;
  if (k >= 2 * M_TOT) return;
  float4 v;
  float* e = (float*)&v;
#pragma unroll
  for (int j = 0; j < 4; ++j) {
    const int kk = k + j;
    int val;
    if (kk < M_TOT) {
      // rows: value = b*128 + i == kk >> 7   (kk = b*16384 + i*128 + j)
      val = kk >> 7;
    } else {
      // cols: value = b*128 + j
      const int kc = kk - M_TOT;
      val = ((kc >> 14) << 7) + (kc & (N_PER - 1));
    }
    e[j] = (float)val;     // single v_cvt_f32_i32
  }
  ((float4*)out)[k4] = v;
}

// ---------------------------------------------------------------------------
// Kernel 2: one workgroup per (graph, 4-row block).
//   - zero 128KB LDS tile (ds_store_b128)
//   - scan the graph's 2048 edges; matching edges ds_add_f32 their 64-f32
//     attr row into the tile (one edge per wave, lanes cover D as float2)
//   - add identity_bias on the diagonal entries of the block
//   - TDM tensor_store_from_lds streams the finished tile to HBM; the
//     co-resident second workgroup on the WGP overlaps its scatter with
//     this group's store drain.
// ---------------------------------------------------------------------------
extern "C" __global__ __launch_bounds__(512)
void scatter_tile_kernel(const int*   __restrict__ edge_index,  // [2, E]
                         const float* __restrict__ edge_attr,   // [E, 64]
                         const float* __restrict__ bias,        // [64]
                         float*       __restrict__ attr_out) {  // [B*N*N, 64]
  extern __shared__ float tile[];   // TILE_FLOATS f32 at LDS offset 0
  const int tid = threadIdx.x;
  const int g   = blockIdx.x / BLKS_PER_G;
  const int rb  = blockIdx.x % BLKS_PER_G;
  const int r0  = rb * ROWS_PER_BLK;

  // ---- zero the tile ----
  float4* t4 = (float4*)tile;
  const float4 z4 = make_float4(0.f, 0.f, 0.f, 0.f);
  for (int i = tid; i < TILE_FLOATS / 4; i += blockDim.x) t4[i] = z4;
  __syncthreads();

  // ---- scatter-sum edges into the tile via LDS f32 atomics ----
  const int wave   = tid >> 5;
  const int lane   = tid & 31;
  const int nwaves = blockDim.x >> 5;
  const int ebase  = g * EPG;
  for (int e = wave; e < EPG; e += nwaves) {
    const int eid = ebase + e;
    const int src = edge_index[eid];
    const int dst = edge_index[E_TOT + eid];
    const int rl  = (src & (N_PER - 1)) - r0;      // local row in this block
    if ((unsigned)rl < (unsigned)ROWS_PER_BLK) {
      const int c = dst & (N_PER - 1);
      const float2 v =
          ((const float2*)edge_attr)[(long long)eid * (D_DIM / 2) + lane];
      float* p = &tile[((rl * N_PER + c) << 6) + (lane << 1)];
      atomicAdd(p, v.x);        // ds_add_f32 (non-returning)
      atomicAdd(p + 1, v.y);
    }
  }
  __syncthreads();

  // ---- identity bias on the diagonal rows of this block ----
  if (tid < ROWS_PER_BLK * D_DIM) {
    const int j = tid >> 6;            // local row 0..3
    const int d = tid & (D_DIM - 1);
    const int i = r0 + j;              // diagonal column == row index in graph
    tile[((j * N_PER + i) << 6) + d] += bias[d];
  }
  __syncthreads();

  // ---- stream the 128KB tile to HBM via the Tensor Data Mover ----
  const uint64_t gaddr = (uint64_t)(uintptr_t)attr_out +
                         (uint64_t)blockIdx.x * TILE_FLOATS * sizeof(float);
#if __has_builtin(__builtin_amdgcn_tensor_store_from_lds) && \
    __has_builtin(__builtin_amdgcn_s_wait_tensorcnt)
  if (tid < 32) {   // one wave issues the TDM descriptor (EXEC-independent op)
    u32x4 g0;
    g0[0] = 1u;                                   // count=1 (valid), user mode
    g0[1] = 0u;                                   // lds_addr = 0 (tile at base)
    g0[2] = (uint32_t)(gaddr & 0xFFFFFFFFull);    // global_addr[31:0]
    g0[3] = ((uint32_t)(gaddr >> 32) & 0x1FFFFFFu) | (2u << 30);  // hi, type=2
    i32x8 g1;
    g1[0] = 0x00020000;                              // data_size = 4 bytes
    g1[1] = (int)((uint32_t)(N_PER * D_DIM) << 16);  // tensor_dim0 = 8192
    g1[2] = (int)((uint32_t)ROWS_PER_BLK << 16);     // tensor_dim1 = 4
    g1[3] = (int)((uint32_t)(N_PER * D_DIM) << 16);  // tile_dim0   = 8192
    g1[4] = ROWS_PER_BLK;                            // tile_dim1   = 4
    g1[5] = N_PER * D_DIM;                           // tensor_dim0_stride
    g1[6] = 0;
    g1[7] = 0;
    i32x4 gz = {0, 0, 0, 0};
#if defined(__clang_major__) && (__clang_major__ >= 23)
    i32x8 gz8 = {0, 0, 0, 0, 0, 0, 0, 0};
    __builtin_amdgcn_tensor_store_from_lds(g0, g1, gz, gz, gz8, 0);
#else
    __builtin_amdgcn_tensor_store_from_lds(g0, g1, gz, gz, 0);
#endif
    __builtin_amdgcn_s_wait_tensorcnt(0);
  }
#else
  // Fallback: wide vector stores from LDS through VGPRs
  float4* o4 = (float4*)((char*)attr_out +
                         (size_t)blockIdx.x * TILE_FLOATS * sizeof(float));
  for (int i = tid; i < TILE_FLOATS / 4; i += blockDim.x) o4[i] = t4[i];
#endif
}

// ---------------------------------------------------------------------------
extern "C" void kernel_launch(void* const* d_in, const int* in_sizes, int n_in,
                              void* d_out, int out_size, void* d_ws, size_t ws_size,
                              hipStream_t stream) {
  const int*   edge_index = (const int*)d_in[0];    // [2, E] int32
  const float* edge_attr  = (const float*)d_in[1];  // [E, 64] f32
  const float* bias       = (const float*)d_in[2];  // [1, 64] f32
  float* out = (float*)d_out;

  // Index outputs: 2*M values, float4-vectorized
  {
    const int total4 = (2 * M_TOT) / 4;               // 1,048,576 float4s
    const int threads = 256;
    const int blocks = (total4 + threads - 1) / threads;
    idx_fill_kernel<<<blocks, threads, 0, stream>>>(out);
  }
  // Dense attr tiles: B*32 workgroups, 512 threads, 128KB dynamic LDS each
  // (two workgroups co-resident per 320KB WGP -> scatter overlaps TDM drain)
  {
    dim3 grid(B_G * BLKS_PER_G);
    scatter_tile_kernel<<<grid, 512, TILE_FLOATS * sizeof(float), stream>>>(
        edge_index, edge_attr, bias, out + 2 * M_TOT);
  }
}